// TemporalAttentionLayer_70059506532938
// MI455X (gfx1250) — compile-verified
//
#include <hip/hip_runtime.h>
#include <hip/hip_bf16.h>

#define B_  32
#define K_  1024
#define F_  128
#define E_  128
#define O2_ 125
#define ALPHA 0.2f

typedef __attribute__((ext_vector_type(2))) float v2f;
typedef __attribute__((ext_vector_type(8))) float v8f;

// ws layout (floats)
#define W1_OFF 0
#define W2_OFF 128
#define C_OFF  256          // [256]=c1 [257]=c2
#define S1_OFF 512
#define S2_OFF (512 + B_*K_)
#define H_OFF  (512 + 2*B_*K_)

// LDS padded strides (dwords) - chosen for bank-conflict-free WMMA frag reads
#define AT_S 66             // A tile row stride: lane m -> banks {2m,2m+1}
#define XT_S 136            // X tile row stride: rows r and r+2 windows disjoint mod 64

// ---------------- Kernel 1: w1 = lin_w^T a1, w2 = lin_w^T a2, c1,c2 ----------
__global__ void prep_kernel(const float* __restrict__ lin_w,
                            const float* __restrict__ lin_b,
                            const float* __restrict__ a,
                            float* __restrict__ ws) {
    int f = threadIdx.x;  // 128 threads
    float acc1 = 0.f, acc2 = 0.f;
    for (int e = 0; e < E_; ++e) {
        float w = lin_w[e * F_ + f];
        acc1 += w * a[e];
        acc2 += w * a[E_ + e];
    }
    ws[W1_OFF + f] = acc1;
    ws[W2_OFF + f] = acc2;
    if (f == 0) {
        float c1 = 0.f, c2 = 0.f;
        for (int e = 0; e < E_; ++e) {
            c1 += lin_b[e] * a[e];
            c2 += lin_b[e] * a[E_ + e];
        }
        ws[C_OFF + 0] = c1;
        ws[C_OFF + 1] = c2;
    }
}

// ---------------- Kernel 2: s1[b,k] = x[b,k,:]·w1 + c1 (one wave per row) ----
__global__ void scores_kernel(const float* __restrict__ x,
                              float* __restrict__ ws) {
    const float* w1 = ws + W1_OFF;
    const float* w2 = ws + W2_OFF;
    int lane = threadIdx.x & 31;
    int wid  = threadIdx.x >> 5;
    int row  = blockIdx.x * 8 + wid;     // row in [0, B*K)
    const float* xr = x + (size_t)row * F_;
    float a1 = 0.f, a2 = 0.f;
#pragma unroll
    for (int i = 0; i < 4; ++i) {
        int f = lane + 32 * i;
        float xv = xr[f];
        a1 += xv * w1[f];
        a2 += xv * w2[f];
    }
#pragma unroll
    for (int off = 16; off >= 1; off >>= 1) {
        a1 += __shfl_xor(a1, off, 32);
        a2 += __shfl_xor(a2, off, 32);
    }
    if (lane == 0) {
        ws[S1_OFF + row] = a1 + ws[C_OFF + 0];
        ws[S2_OFF + row] = a2 + ws[C_OFF + 1];
    }
}

// ---------------- Kernel 3: softmax row -> attention (into d_out) ------------
__global__ __launch_bounds__(256)
void softmax_kernel(const float* __restrict__ bias,
                    const float* __restrict__ ws,
                    float* __restrict__ att_out) {
    __shared__ float sred[8];
    int i = blockIdx.x;          // row within batch
    int b = blockIdx.y;          // batch
    int tid  = threadIdx.x;
    int lane = tid & 31;
    int wid  = tid >> 5;

    float s1i = ws[S1_OFF + b * K_ + i];
    const float* s2r = ws + S2_OFF + b * K_;
    const float* br  = bias + (size_t)i * K_;

    float ev[4];
    float m = -3.4e38f;
#pragma unroll
    for (int t = 0; t < 4; ++t) {
        int j = tid + t * 256;
        float e = s1i + s2r[j];
        e = (e > 0.f) ? e : ALPHA * e;
        e += br[j];
        ev[t] = e;
        m = fmaxf(m, e);
    }
#pragma unroll
    for (int off = 16; off >= 1; off >>= 1) m = fmaxf(m, __shfl_xor(m, off, 32));
    if (lane == 0) sred[wid] = m;
    __syncthreads();
    if (tid < 8) {
        float t = sred[tid];
#pragma unroll
        for (int off = 4; off >= 1; off >>= 1) t = fmaxf(t, __shfl_xor(t, off, 32));
        if (tid == 0) sred[0] = t;
    }
    __syncthreads();
    float rowmax = sred[0];
    __syncthreads();

    float s = 0.f;
#pragma unroll
    for (int t = 0; t < 4; ++t) {
        ev[t] = __expf(ev[t] - rowmax);
        s += ev[t];
    }
#pragma unroll
    for (int off = 16; off >= 1; off >>= 1) s += __shfl_xor(s, off, 32);
    if (lane == 0) sred[wid] = s;
    __syncthreads();
    if (tid < 8) {
        float t = sred[tid];
#pragma unroll
        for (int off = 4; off >= 1; off >>= 1) t += __shfl_xor(t, off, 32);
        if (tid == 0) sred[0] = t;
    }
    __syncthreads();
    float inv = 1.0f / sred[0];

    float* ar = att_out + ((size_t)b * K_ + i) * K_;
#pragma unroll
    for (int t = 0; t < 4; ++t) {
        int j = tid + t * 256;
        ar[j] = ev[t] * inv;
    }
}

// ---------------- Kernel 4: h = attention @ x + x  (WMMA f32 16x16x4) --------
// grid: (K/64, B). block: 256 threads (8 waves). Block computes a 64x128
// output slab; waves form a 2(M-group) x 4(N-group) grid, each wave owning a
// 2x2 grid of 16x16 tiles (32x32) so every A/B fragment feeds 2 WMMAs.
// A slab (64x64 att) and X slab (64x128) staged in LDS with padded strides.
__global__ __launch_bounds__(256)
void attnx_kernel(const float* __restrict__ att,
                  const float* __restrict__ x,
                  float* __restrict__ h) {
    __shared__ float At[64 * AT_S];   // ~16.5 KB
    __shared__ float Xt[64 * XT_S];   // ~34.0 KB

    int mblk = blockIdx.x * 64;       // output row base
    int b    = blockIdx.y;
    int tid  = threadIdx.x;
    int lane = tid & 31;
    int wid  = tid >> 5;

    int mg = wid >> 2;                // M-group (0..1)  -> rows  mg*32 .. +31
    int ng = wid & 3;                 // N-group (0..3)  -> cols  ng*32 .. +31
    int am0 = mg * 32;
    int bn0 = ng * 32;

    int m  = lane & 15;               // row within 16-tile (A) / col (B,D)
    int kq = (lane >> 4) * 2;         // K sub-offset: 0 (lanes 0-15), 2 (16-31)
    int n  = lane & 15;

    const float* Ab = att + ((size_t)b * K_ + (size_t)mblk) * K_;
    const float* Xb = x + (size_t)b * K_ * F_;

    v8f acc[2][2] = {};

    for (int kb = 0; kb < K_; kb += 64) {
        // --- cooperative loads (float4, coalesced, padded LDS rows) ---
#pragma unroll
        for (int q = 0; q < 4; ++q) {  // At: 64x64 = 1024 float4, 4 per thread
            int idx = (tid + q * 256) * 4;
            int r = idx >> 6, c = idx & 63;
            *(float4*)&At[r * AT_S + c] = *(const float4*)&Ab[(size_t)r * K_ + kb + c];
        }
#pragma unroll
        for (int q = 0; q < 8; ++q) {  // Xt: 64x128 = 2048 float4, 8 per thread
            int idx = (tid + q * 256) * 4;
            int r = idx >> 7, c = idx & 127;
            *(float4*)&Xt[r * XT_S + c] = *(const float4*)&Xb[(size_t)(kb + r) * F_ + c];
        }
        __syncthreads();

        // --- 64 WMMAs over this 64-deep slab (4 tiles x 16 k-steps) ---
#pragma unroll
        for (int kk = 0; kk < 64; kk += 4) {
            int ka = kk + kq;
            v2f a0, a1, b0, b1;
            {
                float2 t0 = *(const float2*)&At[(am0 + m) * AT_S + ka];
                a0.x = t0.x; a0.y = t0.y;
                float2 t1 = *(const float2*)&At[(am0 + 16 + m) * AT_S + ka];
                a1.x = t1.x; a1.y = t1.y;
            }
            b0.x = Xt[(ka + 0) * XT_S + bn0 + n];
            b0.y = Xt[(ka + 1) * XT_S + bn0 + n];
            b1.x = Xt[(ka + 0) * XT_S + bn0 + 16 + n];
            b1.y = Xt[(ka + 1) * XT_S + bn0 + 16 + n];

            acc[0][0] = __builtin_amdgcn_wmma_f32_16x16x4_f32(
                false, a0, false, b0, (short)0, acc[0][0], false, false);
            acc[0][1] = __builtin_amdgcn_wmma_f32_16x16x4_f32(
                false, a0, false, b1, (short)0, acc[0][1], false, false);
            acc[1][0] = __builtin_amdgcn_wmma_f32_16x16x4_f32(
                false, a1, false, b0, (short)0, acc[1][0], false, false);
            acc[1][1] = __builtin_amdgcn_wmma_f32_16x16x4_f32(
                false, a1, false, b1, (short)0, acc[1][1], false, false);
        }
        __syncthreads();
    }

    // --- store with residual: h = acc + x ---
#pragma unroll
    for (int ti = 0; ti < 2; ++ti) {
#pragma unroll
        for (int tj = 0; tj < 2; ++tj) {
#pragma unroll
            for (int r = 0; r < 8; ++r) {
                int row = mblk + am0 + ti * 16 + r + ((lane < 16) ? 0 : 8);
                int col = bn0 + tj * 16 + n;
                size_t gi = (size_t)b * K_ * F_ + (size_t)row * F_ + col;
                h[gi] = acc[ti][tj][r] + x[gi];
            }
        }
    }
}

// ---------------- Kernel 5: h2[b,f,o] = relu(sum_k h[b,k,f]*w2[o,k] + b2[o]) -
__global__ __launch_bounds__(128)
void out2_kernel(const float* __restrict__ h,
                 const float* __restrict__ lin2_w,
                 const float* __restrict__ lin2_b,
                 float* __restrict__ h2) {
    int o = blockIdx.x;
    int b = blockIdx.y;
    int f = threadIdx.x;
    const float* hb = h + (size_t)b * K_ * F_ + f;
    const float* wr = lin2_w + (size_t)o * K_;
    float acc = lin2_b[o];
#pragma unroll 4
    for (int k = 0; k < K_; ++k) {
        acc += hb[(size_t)k * F_] * wr[k];
    }
    acc = fmaxf(acc, 0.f);
    h2[(size_t)b * F_ * O2_ + (size_t)f * O2_ + o] = acc;
}

extern "C" void kernel_launch(void* const* d_in, const int* in_sizes, int n_in,
                              void* d_out, int out_size, void* d_ws, size_t ws_size,
                              hipStream_t stream) {
    const float* x      = (const float*)d_in[0];
    const float* lin_w  = (const float*)d_in[1];
    const float* lin_b  = (const float*)d_in[2];
    const float* a      = (const float*)d_in[3];
    const float* bias   = (const float*)d_in[4];
    const float* lin2_w = (const float*)d_in[5];
    const float* lin2_b = (const float*)d_in[6];

    float* out   = (float*)d_out;
    float* h2    = out;                                  // (B, F, O2)
    float* attn  = out + (size_t)B_ * F_ * O2_;          // (B, K, K)
    float* ws    = (float*)d_ws;
    float* hbuf  = ws + H_OFF;

    prep_kernel<<<1, 128, 0, stream>>>(lin_w, lin_b, a, ws);

    scores_kernel<<<(B_ * K_) / 8, 256, 0, stream>>>(x, ws);

    softmax_kernel<<<dim3(K_, B_), 256, 0, stream>>>(bias, ws, attn);

    attnx_kernel<<<dim3(K_ / 64, B_), 256, 0, stream>>>(attn, x, hbuf);

    out2_kernel<<<dim3(O2_, B_), 128, 0, stream>>>(hbuf, lin2_w, lin2_b, h2);
}